// SSVDModel_53901839565583
// MI455X (gfx1250) — compile-verified
//
#include <hip/hip_runtime.h>
#include <hip/hip_bf16.h>

typedef __attribute__((ext_vector_type(2))) float v2f;
typedef __attribute__((ext_vector_type(8))) float v8f;
typedef __attribute__((ext_vector_type(4))) unsigned int u32x4;
typedef __attribute__((ext_vector_type(8))) int i32x8;
typedef __attribute__((ext_vector_type(4))) int i32x4;

#define NPIX (256 * 128 * 128)

#if defined(__has_builtin)
#if __has_builtin(__builtin_amdgcn_tensor_load_to_lds) && \
    __has_builtin(__builtin_amdgcn_s_wait_tensorcnt)
#define HAVE_TDM 1
#endif
#endif

// ---------------------------------------------------------------------------
// Shared-memory layout for the SVD+MLP kernel (float offsets)
// ---------------------------------------------------------------------------
#define WC_OFF   0            // 128 cols x 129  (A working copy, col-major)
#define VC_OFF   16512        // 128 cols x 129  (V accumulator, col-major)
#define RA_OFF   33024        // 128 x 66 activations ping (also TDM stage, 128x129)
#define RB_OFF   41472        // 128 x 66 activations pong
#define VHT_OFF  49920        // 128 x 66  Vh_st stored transposed (N-major)
#define WL_OFF   58368        // 64 x 66   staged layer weight, transposed (N-major)
#define KEY_OFF  62592        // 128 sort keys
#define IDX_OFF  62720        // 128 sort indices (int)
#define SV_OFF   62848        // 64 singular values
#define SGU_OFF  62912        // 64 U column signs
#define SGV_OFF  62976        // 64 V column signs
#define SMEM_FLOATS 63040
#define SMEM_BYTES  (SMEM_FLOATS * 4)

// ---------------------------------------------------------------------------
// Kernel 1: feature extraction + 1D conv chain  ->  M[b,h,w]  (memory bound)
// ---------------------------------------------------------------------------
__global__ __launch_bounds__(256) void k_featconv(
    const float* __restrict__ obs,
    const float* __restrict__ c1w, const float* __restrict__ c1b,
    const float* __restrict__ c2w, const float* __restrict__ c2b,
    float* __restrict__ M)
{
    int pix = blockIdx.x * 256 + threadIdx.x;
    if (pix >= NPIX) return;

    const float* o = obs + (size_t)pix * 29;
    float ov[29];
#pragma unroll
    for (int d = 0; d < 29; ++d) ov[d] = o[d];

    // feats[f] = sum_d arange(size)[d] * obs[..., off+d]
    const int off[6] = {0, 5, 10, 13, 21, 27};
    const int sz[6]  = {5, 5, 3, 8, 6, 2};
    float x0[6];
#pragma unroll
    for (int f = 0; f < 6; ++f) {
        float acc = 0.f;
#pragma unroll
        for (int d = 1; d < 8; ++d)
            if (d < sz[f]) acc += ov[off[f] + d] * (float)d;
        x0[f] = acc;
    }

    float w4[4] = {c1w[0], c1w[1], c1w[2], c1w[3]};
    float b1 = c1b[0];

    // conv k=4, stride=2, pad=2 (correlation, zero-padded):  6 -> 4 -> 3 -> 2
    float x1[4];
#pragma unroll
    for (int i = 0; i < 4; ++i) {
        float acc = b1;
#pragma unroll
        for (int t = 0; t < 4; ++t) {
            int j = 2 * i - 2 + t;
            if (j >= 0 && j < 6) acc += x0[j] * w4[t];
        }
        x1[i] = acc;
    }
    float x2[3];
#pragma unroll
    for (int i = 0; i < 3; ++i) {
        float acc = b1;
#pragma unroll
        for (int t = 0; t < 4; ++t) {
            int j = 2 * i - 2 + t;
            if (j >= 0 && j < 4) acc += x1[j] * w4[t];
        }
        x2[i] = acc;
    }
    float x3[2];
#pragma unroll
    for (int i = 0; i < 2; ++i) {
        float acc = b1;
#pragma unroll
        for (int t = 0; t < 4; ++t) {
            int j = 2 * i - 2 + t;
            if (j >= 0 && j < 3) acc += x2[j] * w4[t];
        }
        x3[i] = acc;
    }
    // conv k=2, stride=1, pad=0: 2 -> 1
    M[pix] = x3[0] * c2w[0] + x3[1] * c2w[1] + c2b[0];
}

// ---------------------------------------------------------------------------
// f32 WMMA 16x16x4 tile step (wave32; EXEC all ones at call sites)
// ---------------------------------------------------------------------------
__device__ __forceinline__ v8f wmma_f32_step(v2f a, v2f b, v8f c) {
    return __builtin_amdgcn_wmma_f32_16x16x4_f32(
        /*neg_a=*/false, a, /*neg_b=*/false, b,
        /*c_mod=*/(short)0, c, /*reuse_a=*/false, /*reuse_b=*/false);
}

// ---------------------------------------------------------------------------
// Kernel 2: per-batch one-sided Jacobi SVD (in LDS) + WMMA-f32 MLP chain
// One workgroup (256 threads = 8 waves) per batch element.
// M tile is staged global->LDS via the Tensor Data Mover when available.
// ---------------------------------------------------------------------------
__global__ __launch_bounds__(256) void k_svdmlp(
    const float* __restrict__ M,
    const float* __restrict__ W1,   // (4,64,64)
    const float* __restrict__ W2,   // (4,64,64)
    float* __restrict__ Yst)        // (256, 16384)
{
    extern __shared__ float sm[];
    float* Wc   = sm + WC_OFF;
    float* Vc   = sm + VC_OFF;
    float* rA   = sm + RA_OFF;
    float* rB   = sm + RB_OFF;
    float* Vht  = sm + VHT_OFF;
    float* Wl   = sm + WL_OFF;
    float* key  = sm + KEY_OFF;
    int*   sidx = (int*)(sm + IDX_OFF);
    float* sval = sm + SV_OFF;
    float* sgu  = sm + SGU_OFF;
    float* sgv  = sm + SGV_OFF;

    const int tid  = threadIdx.x;
    const int lane = tid & 31;
    const int wave = tid >> 5;
    const int a    = blockIdx.x;

    const float* Ma = M + (size_t)a * 16384;

#ifdef HAVE_TDM
    // ---- TDM: DMA the 128x128 f32 tile into LDS (row stride padded to 129
    //      DWORDs via pad_interval=128 DW, pad_amount=1 DW), wave 0 issues.
    float* Mst = sm + RA_OFF;           // 128 x 129 staging (reused by rA later)
    if (wave == 0) {
        unsigned long long ga = (unsigned long long)(uintptr_t)Ma;
        u32x4 g0 = {
            1u,                                   // count=1 (valid user D#)
            (unsigned)(RA_OFF * 4),               // lds_addr (bytes)
            (unsigned)(ga & 0xFFFFFFFFull),       // global_addr[31:0]
            (unsigned)((ga >> 32) & 0x1FFFFFFull) | (2u << 30) // addr[56:32]|type=2
        };
        i32x8 g1 = {
            (int)((2u << 16) | (1u << 20) | (6u << 22)), // data_size=4B, pad_en, interval=128DW
            (int)(128u << 16),                    // tensor_dim0[15:0]=128
            (int)(128u << 16),                    // tensor_dim1[15:0]=128
            (int)(128u << 16),                    // tile_dim0=128
            128,                                  // tile_dim1=128, tile_dim2=0
            128,                                  // tensor_dim0_stride[31:0]=128
            (int)(16384u << 16),                  // tensor_dim1_stride[15:0]
            0
        };
        i32x4 gz4 = {0, 0, 0, 0};
        i32x8 gz8 = {0, 0, 0, 0, 0, 0, 0, 0};
        __builtin_amdgcn_tensor_load_to_lds(g0, g1, gz4, gz4, gz8, 0);
        __builtin_amdgcn_s_wait_tensorcnt(0);
    }
    for (int i = tid; i < 16384; i += 256) {      // V = I meanwhile
        int row = i >> 7, col = i & 127;
        Vc[col * 129 + row] = (row == col) ? 1.f : 0.f;
    }
    __syncthreads();
    for (int i = tid; i < 16384; i += 256) {      // transpose stage -> Wc
        int row = i >> 7, col = i & 127;
        Wc[col * 129 + row] = Mst[row * 129 + col];
    }
    __syncthreads();
#else
    for (int i = tid; i < 16384; i += 256) {
        int row = i >> 7, col = i & 127;
        Wc[col * 129 + row] = Ma[i];
        Vc[col * 129 + row] = (row == col) ? 1.f : 0.f;
    }
    __syncthreads();
#endif

    // ---- one-sided Jacobi: 12 sweeps x 127 rounds x 64 disjoint pairs ---
    const int gp = tid >> 2;   // pair id 0..63 (4 lanes per pair)
    const int gl = tid & 3;
    for (int sweep = 0; sweep < 12; ++sweep) {
        for (int rr = 0; rr < 127; ++rr) {
            int p, q;
            if (gp == 0) { p = 127; q = rr; }
            else { p = (rr + gp) % 127; q = (rr - gp + 127) % 127; }
            if (p > q) { int t = p; p = q; q = t; }
            float* wp = Wc + p * 129; float* wq = Wc + q * 129;
            float* vp = Vc + p * 129; float* vq = Vc + q * 129;

            float al = 0.f, be = 0.f, ga = 0.f;
            for (int r = gl; r < 128; r += 4) {
                float x = wp[r], y = wq[r];
                al += x * x; be += y * y; ga += x * y;
            }
            al += __shfl_xor(al, 1); al += __shfl_xor(al, 2);
            be += __shfl_xor(be, 1); be += __shfl_xor(be, 2);
            ga += __shfl_xor(ga, 1); ga += __shfl_xor(ga, 2);

            float c = 1.f, s = 0.f;
            if (fabsf(ga) > 1e-12f * sqrtf(al * be) + 1e-30f) {
                float zeta = (be - al) / (2.f * ga);
                float t = copysignf(1.f, zeta) /
                          (fabsf(zeta) + sqrtf(1.f + zeta * zeta));
                c = rsqrtf(1.f + t * t);
                s = c * t;
            }
            for (int r = gl; r < 128; r += 4) {
                float x = wp[r], y = wq[r];
                wp[r] = c * x - s * y;
                wq[r] = s * x + c * y;
                float vx = vp[r], vy = vq[r];
                vp[r] = c * vx - s * vy;
                vq[r] = s * vx + c * vy;
            }
            __syncthreads();
        }
    }

    // ---- column norms -> sort keys --------------------------------------
    if (tid < 128) {
        const float* wcol = Wc + tid * 129;
        float n2 = 0.f;
        for (int r = 0; r < 128; ++r) n2 += wcol[r] * wcol[r];
        key[tid] = n2;
        sidx[tid] = tid;
    }
    __syncthreads();

    // ---- bitonic sort, descending ---------------------------------------
    for (int k = 2; k <= 128; k <<= 1) {
        for (int j = k >> 1; j > 0; j >>= 1) {
            if (tid < 128) {
                int ixj = tid ^ j;
                if (ixj > tid) {
                    bool descBlk = ((tid & k) == 0);
                    float k0 = key[tid], k1 = key[ixj];
                    bool sw = descBlk ? (k0 < k1) : (k0 > k1);
                    if (sw) {
                        key[tid] = k1; key[ixj] = k0;
                        int t = sidx[tid]; sidx[tid] = sidx[ixj]; sidx[ixj] = t;
                    }
                }
            }
            __syncthreads();
        }
    }

    // ---- singular values + sign standardization (stand-in for QR) -------
    if (tid < 64) {
        int cs = sidx[tid];
        sval[tid] = sqrtf(key[tid]);
        sgu[tid] = (Wc[cs * 129 + 0] < 0.f) ? -1.f : 1.f;
        sgv[tid] = (Vc[cs * 129 + 0] < 0.f) ? -1.f : 1.f;
    }
    __syncthreads();

    // ---- build U_st (into rA, 128x66) and Vh_st transposed (128x66) -----
    for (int i = tid; i < 8192; i += 256) {
        int jc = i >> 7, row = i & 127;
        int cs = sidx[jc];
        float sv = sval[jc];
        float inv = (sv > 1e-20f) ? (sgu[jc] / sv) : 0.f;
        rA[row * 66 + jc]  = Wc[cs * 129 + row] * inv;
        Vht[row * 66 + jc] = sgv[jc] * Vc[cs * 129 + row];   // Vht[n][j] = Vh[j][n]
    }

    // ---- MLP: 7 x (128x64 @ 64x64) with ReLU, Sigma folded into layer 3 -
    const float* layerW[7] = { W1, W1 + 4096, W1 + 8192, W1 + 12288,
                               W2 + 4096, W2 + 8192, W2 + 12288 };
    float* bufs[2] = { rA, rB };
    const int arow = 16 * wave + (lane & 15);
    const int kadd = (lane >> 4) << 1;

    for (int L = 0; L < 7; ++L) {
        const float* Wg = layerW[L];
        const float* Lin = bufs[L & 1];
        float* Lout = bufs[(L + 1) & 1];
        __syncthreads();                         // Wl reads + prev Lout done
        // stage weight transposed: Wl[n][k] so B-pair (k,k+1) is contiguous
        for (int i = tid; i < 4096; i += 256) {
            int k = i >> 6, n = i & 63;
            Wl[n * 66 + k] = Wg[i];
        }
        __syncthreads();

        const bool doScale = (L == 3);           // r = relu(r @ Sigma) fold
        for (int nt = 0; nt < 4; ++nt) {
            v8f acc = {0.f, 0.f, 0.f, 0.f, 0.f, 0.f, 0.f, 0.f};
            const int col = 16 * nt + (lane & 15);
#pragma unroll
            for (int k = 0; k < 64; k += 4) {
                int kb = k + kadd;
                v2f av = *(const v2f*)(Lin + arow * 66 + kb);  // ds_load_b64
                v2f bv = *(const v2f*)(Wl + col * 66 + kb);    // ds_load_b64
                acc = wmma_f32_step(av, bv, acc);
            }
            float scale = doScale ? sval[col] : 1.f;
#pragma unroll
            for (int i = 0; i < 8; ++i) {
                int row = 16 * wave + i + 8 * (lane >> 4);
                Lout[row * 66 + col] = fmaxf(acc[i], 0.f) * scale;
            }
        }
    }
    __syncthreads();

    // ---- Y = r @ Vh_st  (128x64 @ 64x128) -> global scratch --------------
    float* Ya = Yst + (size_t)a * 16384;
    for (int nt = 0; nt < 8; ++nt) {
        v8f acc = {0.f, 0.f, 0.f, 0.f, 0.f, 0.f, 0.f, 0.f};
        const int col = 16 * nt + (lane & 15);
#pragma unroll
        for (int k = 0; k < 64; k += 4) {
            int kb = k + kadd;
            v2f av = *(const v2f*)(rB + arow * 66 + kb);
            v2f bv = *(const v2f*)(Vht + col * 66 + kb);
            acc = wmma_f32_step(av, bv, acc);
        }
#pragma unroll
        for (int i = 0; i < 8; ++i) {
            int row = 16 * wave + i + 8 * (lane >> 4);
            Ya[row * 128 + col] = acc[i];
        }
    }
}

// ---------------------------------------------------------------------------
// Kernel 3: Out(256x128) = Ystack(256x16384) @ Wo^T   (K = 16384, WMMA f32)
// 16 blocks (one per 16-row a-tile), 8 waves = 8 j-tiles per block.
// ---------------------------------------------------------------------------
__global__ __launch_bounds__(256) void k_outgemm(
    const float* __restrict__ Yst,
    const float* __restrict__ Wo,    // (128, 16384) row-major
    float* __restrict__ Out)         // (256, 128)
{
    __shared__ float Yt[16 * 258];
    const int tid  = threadIdx.x;
    const int lane = tid & 31;
    const int wave = tid >> 5;
    const int a0   = blockIdx.x * 16;
    const int jj   = 16 * wave + (lane & 15);
    const int kadd = (lane >> 4) << 1;
    const int arow = lane & 15;
    const float* WoRow = Wo + (size_t)jj * 16384;

    v8f acc = {0.f, 0.f, 0.f, 0.f, 0.f, 0.f, 0.f, 0.f};

    for (int k0 = 0; k0 < 16384; k0 += 256) {
        __syncthreads();
        for (int i = tid; i < 16 * 256; i += 256) {
            int ar = i >> 8, c = i & 255;
            Yt[ar * 258 + c] = Yst[(size_t)(a0 + ar) * 16384 + k0 + c];
        }
        __syncthreads();
        if (k0 + 256 < 16384)
            __builtin_prefetch(&WoRow[k0 + 256], 0, 1);
#pragma unroll 4
        for (int kk = 0; kk < 256; kk += 4) {
            int kb = kk + kadd;
            v2f av = *(const v2f*)(Yt + arow * 258 + kb);     // ds_load_b64
            v2f bv = *(const v2f*)(WoRow + k0 + kb);          // global_load_b64
            acc = wmma_f32_step(av, bv, acc);
        }
    }
#pragma unroll
    for (int i = 0; i < 8; ++i) {
        int aa = a0 + i + 8 * (lane >> 4);
        Out[aa * 128 + jj] = acc[i];
    }
}

// ---------------------------------------------------------------------------
extern "C" void kernel_launch(void* const* d_in, const int* in_sizes, int n_in,
                              void* d_out, int out_size, void* d_ws, size_t ws_size,
                              hipStream_t stream) {
    const float* obs = (const float*)d_in[0];
    const float* w1  = (const float*)d_in[1];
    const float* w2  = (const float*)d_in[2];
    const float* wO  = (const float*)d_in[3];
    const float* c1w = (const float*)d_in[4];
    const float* c1b = (const float*)d_in[5];
    const float* c2w = (const float*)d_in[6];
    const float* c2b = (const float*)d_in[7];

    float* Mbuf = (float*)d_ws;                 // 256*128*128 floats
    float* Ybuf = Mbuf + NPIX;                  // 256*16384 floats

    k_featconv<<<NPIX / 256, 256, 0, stream>>>(obs, c1w, c1b, c2w, c2b, Mbuf);

    (void)hipFuncSetAttribute(reinterpret_cast<const void*>(k_svdmlp),
                              hipFuncAttributeMaxDynamicSharedMemorySize,
                              SMEM_BYTES);
    k_svdmlp<<<256, 256, SMEM_BYTES, stream>>>(Mbuf, w1, w2, Ybuf);

    k_outgemm<<<16, 256, 0, stream>>>(Ybuf, wO, (float*)d_out);
}